// E2EModule_80161269612575
// MI455X (gfx1250) — compile-verified
//
#include <hip/hip_runtime.h>
#include <math.h>

typedef __attribute__((ext_vector_type(2))) float v2f;
typedef __attribute__((ext_vector_type(8))) float v8f;

#define N_OBS 2048
#define N_X   64
#define N_Y   256
#define NITER 200

// LDS pitches chosen for conflict-free access (bank = (row*pitch+col) mod 64):
// PW mod 64 == 4  -> A-fragment b64 reads cover all 64 banks exactly once.
// PS mod 64 == 16 -> B-fragment half-waves (rows k, k+2) use disjoint banks.
#define PS 272   // Sigma pitch   (256 rows * 272 * 4B = 272KB)
#define PW 260   // w/v pitch     (16 rows * 260 * 4B  = 16.25KB each)

#if defined(__has_builtin)
#if __has_builtin(__builtin_amdgcn_global_load_async_to_lds_b128)
#define HAVE_ASYNC_LDS 1
#endif
#endif

#ifdef HAVE_ASYNC_LDS
// Builtin parameter type per hipcc diagnostic: pointer to
// '__attribute__((__vector_size__(4 * sizeof(int)))) int'
typedef __attribute__((__vector_size__(4 * sizeof(int)))) int v4i_async;
typedef __attribute__((address_space(1))) v4i_async* gptr_v4i;
typedef __attribute__((address_space(3))) v4i_async* lptr_v4i;
#endif

__device__ __forceinline__ void wait_async0() {
#if defined(__has_builtin)
#if __has_builtin(__builtin_amdgcn_s_wait_asynccnt)
    __builtin_amdgcn_s_wait_asynccnt(0);
    return;
#endif
#endif
    asm volatile("s_wait_asynccnt 0" ::: "memory");
}

// ---------------------------------------------------------------------------
// Kernel 1: y_hat = x @ W^T + b   via V_WMMA_F32_16X16X4_F32
// grid = (N_Y/16, N_OBS/16), block = 32 (one wave per 16x16 output tile)
// ---------------------------------------------------------------------------
__global__ void yhat_kernel(const float* __restrict__ x,
                            const float* __restrict__ W,
                            const float* __restrict__ bias,
                            float* __restrict__ yhat) {
    const int n0   = blockIdx.x * 16;
    const int m0   = blockIdx.y * 16;
    const int lane = threadIdx.x & 31;
    const int half = lane >> 4;
    const int l    = lane & 15;

    v8f acc;
    const float bv = bias[n0 + l];               // C[m][n] = b[n]
    for (int r = 0; r < 8; ++r) acc[r] = bv;

    for (int k0 = 0; k0 < N_X; k0 += 4) {
        v2f a, b;
        a.x = x[(m0 + l) * N_X + k0 + 2 * half];
        a.y = x[(m0 + l) * N_X + k0 + 2 * half + 1];
        b.x = W[(n0 + l) * N_X + k0 + 2 * half];     // B = W^T
        b.y = W[(n0 + l) * N_X + k0 + 2 * half + 1];
        acc = __builtin_amdgcn_wmma_f32_16x16x4_f32(false, a, false, b,
                                                    (short)0, acc, false, false);
    }
    for (int r = 0; r < 8; ++r)
        yhat[(m0 + r + 8 * half) * N_Y + n0 + l] = acc[r];
}

// ---------------------------------------------------------------------------
// Kernel 2a/2b: column means of (y - y_hat), two-phase deterministic reduction
// ---------------------------------------------------------------------------
__global__ void mean_partial_kernel(const float* __restrict__ y,
                                    const float* __restrict__ yhat,
                                    float* __restrict__ partial) {
    const int n = threadIdx.x;
    const int p = blockIdx.x;                  // 16 blocks, 128 rows each
    float s = 0.0f;
    for (int t = 128 * p; t < 128 * p + 128; ++t)
        s += y[t * N_Y + n] - yhat[t * N_Y + n];
    partial[p * N_Y + n] = s;
}

__global__ void mean_final_kernel(const float* __restrict__ partial,
                                  float* __restrict__ mean) {
    const int n = threadIdx.x;
    float s = 0.0f;
    for (int p = 0; p < 16; ++p) s += partial[p * N_Y + n];
    mean[n] = s * (1.0f / (float)N_OBS);
}

// ---------------------------------------------------------------------------
// Kernel 3: ep_bar = (y - y_hat - mean) / sqrt(n_obs)
// ---------------------------------------------------------------------------
__global__ void ebar_kernel(const float* __restrict__ y,
                            const float* __restrict__ yhat,
                            const float* __restrict__ mean,
                            float* __restrict__ eb) {
    const int idx = blockIdx.x * blockDim.x + threadIdx.x;
    const int n = idx & (N_Y - 1);
    eb[idx] = (y[idx] - yhat[idx] - mean[n]) * 0.022097086912079612f; // 1/sqrt(2048)
}

// ---------------------------------------------------------------------------
// Kernel 4: Sigma = eb^T @ eb  (256x256, K=2048) via fp32 WMMA.
// grid = 256 blocks (16x16 tiles), block = 32.
// ---------------------------------------------------------------------------
__global__ void sigma_kernel(const float* __restrict__ eb,
                             float* __restrict__ Sigma) {
    const int n0   = (blockIdx.x & 15) * 16;
    const int m0   = (blockIdx.x >> 4) * 16;
    const int lane = threadIdx.x & 31;
    const int half = lane >> 4;
    const int l    = lane & 15;

    v8f acc = (v8f){0, 0, 0, 0, 0, 0, 0, 0};
    for (int k0 = 0; k0 < N_OBS; k0 += 4) {
        v2f a, b;
        a.x = eb[(k0 + 2 * half) * N_Y + m0 + l];       // A = eb^T
        a.y = eb[(k0 + 2 * half + 1) * N_Y + m0 + l];
        b.x = eb[(k0 + 2 * half) * N_Y + n0 + l];
        b.y = eb[(k0 + 2 * half + 1) * N_Y + n0 + l];
        acc = __builtin_amdgcn_wmma_f32_16x16x4_f32(false, a, false, b,
                                                    (short)0, acc, false, false);
    }
    for (int r = 0; r < 8; ++r)
        Sigma[(m0 + r + 8 * half) * N_Y + n0 + l] = acc[r];
}

// ---------------------------------------------------------------------------
// Kernel 5: power iteration (Rayleigh quotient) for lambda_max(Sigma).
// 1 block x 256 threads; wave-level shfl reductions, 3 barriers per iteration.
// ---------------------------------------------------------------------------
__global__ void power_kernel(const float* __restrict__ Sigma,
                             float* __restrict__ stepOut) {
    __shared__ float v[N_Y];
    __shared__ float red2[8];
    __shared__ float redq[8];
    const int i    = threadIdx.x;
    const int lane = i & 31;
    const int wv   = i >> 5;
    v[i] = 1.0f + 0.001f * (float)i;   // symmetry-breaking start
    __syncthreads();

    float lam = 0.0f;
    for (int it = 0; it < 128; ++it) {
        float u = 0.0f;
        for (int k = 0; k < N_Y; ++k) u = fmaf(Sigma[i * N_Y + k], v[k], u);
        float r2 = u * u;          // ||u||^2 part
        float rq = v[i] * u;       // Rayleigh numerator part (v normalized)
        for (int m = 16; m > 0; m >>= 1) {
            r2 += __shfl_xor(r2, m, 32);
            rq += __shfl_xor(rq, m, 32);
        }
        if (lane == 0) { red2[wv] = r2; redq[wv] = rq; }
        __syncthreads();
        float nrm2 = 0.0f, ray = 0.0f;
        for (int p = 0; p < 8; ++p) { nrm2 += red2[p]; ray += redq[p]; }
        const float nrm = sqrtf(nrm2);
        __syncthreads();
        v[i] = u / nrm;
        lam = ray;                 // v'Sigma v with v normalized -> lambda_max
        __syncthreads();
    }
    if (i == 0) stepOut[0] = 1.0f / (lam + 1e-8f);
}

// ---------------------------------------------------------------------------
// Kernel 6: batched FISTA, 200 iterations, Sigma resident in LDS.
// grid = 128 WGs (16 rows each), block = 128 (4 waves); wave w computes N-tiles
// [64w,64w+64) of s = w @ Sigma with fp32 WMMA, exchange via LDS, wave w then
// projects full rows 4w..4w+3. LDS: 272KB (Sigma) + 2x16.25KB = ~305KB.
// ---------------------------------------------------------------------------
__global__ __launch_bounds__(128, 1)
void fista_kernel(const float* __restrict__ SigmaG,
                  const float* __restrict__ yhat,
                  const float* __restrict__ stepPtr,
                  float* __restrict__ zout) {
    __shared__ float sS[N_Y * PS];    // Sigma, padded pitch
    __shared__ float sW[16 * PW];     // momentum iterate w (16 rows)
    __shared__ float sV[16 * PW];     // s = w @ Sigma exchange buffer

    const int tid  = threadIdx.x;
    const int lane = tid & 31;
    const int wave = tid >> 5;
    const int half = lane >> 4;
    const int l    = lane & 15;
    const int row0 = blockIdx.x * 16;
    const float step = stepPtr[0];

    // ---- Sigma -> LDS via async copy (ASYNCcnt path), row-wise (padded pitch)
    for (int i = tid; i < (N_Y * N_Y) / 4; i += 128) {
        const int row = i >> 6;
        const int c4  = (i & 63) << 2;
#ifdef HAVE_ASYNC_LDS
        __builtin_amdgcn_global_load_async_to_lds_b128(
            (gptr_v4i)(SigmaG + row * N_Y + c4),
            (lptr_v4i)(&sS[row * PS + c4]),
            0, 0);
#else
        *(float4*)(&sS[row * PS + c4]) = *(const float4*)(SigmaG + row * N_Y + c4);
#endif
    }
    // w0 = z0 = 1/n_y
    for (int i = tid; i < 16 * N_Y; i += 128)
        sW[(i >> 8) * PW + (i & 255)] = 1.0f / (float)N_Y;

    // Per-wave projection-layout state: rows rbase..rbase+3, lane holds col lane+32j
    const int rbase = 4 * wave;
    float z[4][8], c[4][8];
    for (int r = 0; r < 4; ++r)
        for (int j = 0; j < 8; ++j) {
            const int col = lane + 32 * j;
            z[r][j] = 1.0f / (float)N_Y;
            c[r][j] = step * yhat[(row0 + rbase + r) * N_Y + col];
        }
#ifdef HAVE_ASYNC_LDS
    wait_async0();
#endif
    __syncthreads();

    float t = 1.0f;
    const int n0w = wave * 64;  // this wave's 4 N-tiles

    for (int it = 0; it < NITER; ++it) {
        // ---- s = w @ Sigma, this wave's N strip -----------------------------
        v8f acc[4];
        for (int jt = 0; jt < 4; ++jt) acc[jt] = (v8f){0, 0, 0, 0, 0, 0, 0, 0};
        for (int k0 = 0; k0 < N_Y; k0 += 4) {
            v2f a;
            a.x = sW[l * PW + k0 + 2 * half];
            a.y = sW[l * PW + k0 + 2 * half + 1];
            for (int jt = 0; jt < 4; ++jt) {
                const int n0 = n0w + jt * 16;
                v2f b;
                b.x = sS[(k0 + 2 * half) * PS + n0 + l];
                b.y = sS[(k0 + 2 * half + 1) * PS + n0 + l];
                acc[jt] = __builtin_amdgcn_wmma_f32_16x16x4_f32(
                    false, a, false, b, (short)0, acc[jt], false, false);
            }
        }
        for (int jt = 0; jt < 4; ++jt) {
            const int n0 = n0w + jt * 16;
            for (int r = 0; r < 8; ++r)
                sV[(r + 8 * half) * PW + n0 + l] = acc[jt][r];
        }
        __syncthreads();

        const float tn   = 0.5f * (1.0f + sqrtf(1.0f + 4.0f * t * t));
        const float beta = (t - 1.0f) / tn;
        t = tn;

        // ---- projection + momentum on rows rbase..rbase+3 -------------------
        for (int r = 0; r < 4; ++r) {
            float v[8];
            float vmax = -1e30f;
            for (int j = 0; j < 8; ++j) {
                const int idx = (rbase + r) * PW + lane + 32 * j;
                const float val = sW[idx] + c[r][j] - step * sV[idx];
                v[j] = val;
                vmax = fmaxf(vmax, val);
            }
            for (int m = 16; m > 0; m >>= 1)
                vmax = fmaxf(vmax, __shfl_xor(vmax, m, 32));
            // theta in [vmax-1, vmax]; bisection on f(th)=sum(max(v-th,0))-1
            float lo = vmax - 1.0f, hi = vmax;
            for (int bs = 0; bs < 48; ++bs) {
                const float th = 0.5f * (lo + hi);
                float s = 0.0f;
                for (int j = 0; j < 8; ++j) s += fmaxf(v[j] - th, 0.0f);
                for (int m = 16; m > 0; m >>= 1) s += __shfl_xor(s, m, 32);
                if (s > 1.0f) lo = th; else hi = th;
            }
            const float th = 0.5f * (lo + hi);
            for (int j = 0; j < 8; ++j) {
                const int col = lane + 32 * j;
                const float zn = fmaxf(v[j] - th, 0.0f);
                const float wn = zn + beta * (zn - z[r][j]);
                z[r][j] = zn;
                sW[(rbase + r) * PW + col] = wn;
            }
        }
        __syncthreads();
    }

    for (int r = 0; r < 4; ++r)
        for (int j = 0; j < 8; ++j)
            zout[(row0 + rbase + r) * N_Y + lane + 32 * j] = z[r][j];
}

// ---------------------------------------------------------------------------
extern "C" void kernel_launch(void* const* d_in, const int* in_sizes, int n_in,
                              void* d_out, int out_size, void* d_ws, size_t ws_size,
                              hipStream_t stream) {
    const float* x = (const float*)d_in[0];   // [2048,64]
    const float* y = (const float*)d_in[1];   // [2048,256]
    const float* W = (const float*)d_in[2];   // [256,64]
    const float* b = (const float*)d_in[3];   // [256]

    float* out   = (float*)d_out;
    float* zout  = out;                       // [2048*256]
    float* yhat  = out + N_OBS * N_Y;         // [2048*256]

    float* ws      = (float*)d_ws;
    float* eb      = ws;                      // [2048*256]
    float* Sigma   = eb + N_OBS * N_Y;        // [256*256]
    float* mean    = Sigma + N_Y * N_Y;       // [256]
    float* stepB   = mean + N_Y;              // [1]
    float* partial = stepB + 1;               // [16*256]

    yhat_kernel<<<dim3(N_Y / 16, N_OBS / 16), 32, 0, stream>>>(x, W, b, yhat);
    mean_partial_kernel<<<16, N_Y, 0, stream>>>(y, yhat, partial);
    mean_final_kernel<<<1, N_Y, 0, stream>>>(partial, mean);
    ebar_kernel<<<N_OBS, N_Y, 0, stream>>>(y, yhat, mean, eb);
    sigma_kernel<<<(N_Y / 16) * (N_Y / 16), 32, 0, stream>>>(eb, Sigma);
    power_kernel<<<1, N_Y, 0, stream>>>(Sigma, stepB);
    fista_kernel<<<N_OBS / 16, 128, 0, stream>>>(Sigma, yhat, stepB, zout);
}